// EnergyModel_36326833389926
// MI455X (gfx1250) — compile-verified
//
#include <hip/hip_runtime.h>
#include <math.h>

#define NNODES  50000
#define NEDGES  400000
#define NGRAPHS 512
#define FDIM    32
#define CUTOFF_R 5.0f

typedef __attribute__((ext_vector_type(2))) float v2f;
typedef __attribute__((ext_vector_type(8))) float v8f;

// ---------------------------------------------------------------------------
// 16x16 (M,N) x K=32 f32 GEMM tile via 8 chained V_WMMA_F32_16X16X4_F32.
// A: LDS, 16 rows x 32 cols row-major.  W: LDS, 32x32 row-major (K x N).
// A fragment layout (ISA 7.12.2, 32-bit A 16x4): lanes 0-15 hold K=4k..4k+1,
// lanes 16-31 hold K=4k+2..4k+3, M = lane%16.  B mirrored (row striped across
// lanes within a VGPR).  C/D: VGPR r -> M=r (lanes 0-15) / M=8+r (lanes 16-31),
// N = lane%16 (+16 for the second N-tile).
// ---------------------------------------------------------------------------
__device__ __forceinline__ void wmma_k32(const float* __restrict__ A,
                                         const float* __restrict__ W,
                                         v8f& acc0, v8f& acc1) {
    const int lane = threadIdx.x & 31;
    const int row  = lane & 15;
    const int koff = (lane >> 4) << 1;   // 0 for lanes 0-15, 2 for lanes 16-31
    const int col  = lane & 15;
#pragma unroll
    for (int kk = 0; kk < 8; ++kk) {
        const int kb = kk * 4 + koff;
        v2f a;  a.x  = A[row * 32 + kb];          a.y  = A[row * 32 + kb + 1];
        v2f b0; b0.x = W[kb * 32 + col];          b0.y = W[(kb + 1) * 32 + col];
        v2f b1; b1.x = W[kb * 32 + col + 16];     b1.y = W[(kb + 1) * 32 + col + 16];
        acc0 = __builtin_amdgcn_wmma_f32_16x16x4_f32(false, a, false, b0,
                                                     (short)0, acc0, false, false);
        acc1 = __builtin_amdgcn_wmma_f32_16x16x4_f32(false, a, false, b1,
                                                     (short)0, acc1, false, false);
    }
}

__device__ __forceinline__ float siluf(float x) { return x / (1.0f + expf(-x)); }

__device__ __forceinline__ void fatomic_add(float* p, float v) {
    unsafeAtomicAdd(p, v);   // -> global_atomic_add_f32
}

// ---------------------------------------------------------------------------
// init: wcomb = Wy0[0] + Wx0[0]; zero the accumulators (yacc, y0acc, eg are
// laid out contiguously in the workspace).
// ---------------------------------------------------------------------------
__global__ void init_kernel(const float* __restrict__ Wy0,
                            const float* __restrict__ Wx0,
                            float* __restrict__ wcomb,
                            float* __restrict__ zbase, int zcount) {
    int i = blockIdx.x * 256 + threadIdx.x;
    if (i < 1024) wcomb[i] = Wy0[i] + Wx0[i];
    if (i < zcount) zbase[i] = 0.0f;
}

// x0[n,f] = embed[z[n], f]
__global__ void gather_x0(const float* __restrict__ embed,
                          const int* __restrict__ z,
                          float* __restrict__ x0) {
    int i = blockIdx.x * 256 + threadIdx.x;
    if (i < NNODES * FDIM) {
        int n = i >> 5, f = i & 31;
        x0[i] = embed[z[n] * FDIM + f];
    }
}

// ---------------------------------------------------------------------------
// Edge pass (used for both message layers):
//   per edge: r, rbf(32);  w = rbf @ Wg (32x32, WMMA);  yacc[dst] += w * xf[src]
// 8 waves/block, one 16-edge tile per wave.
// ---------------------------------------------------------------------------
__global__ void edge_pass(const float* __restrict__ pos,
                          const int*   __restrict__ src,
                          const int*   __restrict__ dst,
                          const float* __restrict__ xf,   // N x 32 source features
                          const float* __restrict__ Wg,   // 32 x 32 row-major (K x N)
                          float*       __restrict__ yacc) // N x 32 accumulator
{
    __shared__ float sW[1024];
    __shared__ float logb[32];
    __shared__ float sA[8][16 * 32];
    __shared__ int   sSrc[8][16], sDst[8][16];

    const int tid = threadIdx.x;
    for (int i = tid; i < 1024; i += 256) sW[i] = Wg[i];
    if (tid < 32) {
        float kf = (float)tid;   // log C(31,k) = lgamma(32)-lgamma(k+1)-lgamma(32-k)
        logb[tid] = lgammaf(32.0f) - lgammaf(kf + 1.0f) - lgammaf(32.0f - kf);
    }

    const int wave = tid >> 5, lane = tid & 31;
    const int ln = lane & 15, half = lane >> 4;

    const int tiles = (NEDGES + 15) / 16;
    const int wtile = blockIdx.x * 8 + wave;
    const bool tvalid = (wtile < tiles);
    const int wt = tvalid ? wtile : (tiles - 1);

    int e  = wt * 16 + ln;
    int ec = e < NEDGES ? e : (NEDGES - 1);
    const int s = src[ec], d = dst[ec];
    if (half == 0) { sSrc[wave][ln] = s; sDst[wave][ln] = d; }

    const float dx = pos[s * 3 + 0] - pos[d * 3 + 0];
    const float dy = pos[s * 3 + 1] - pos[d * 3 + 1];
    const float dz = pos[s * 3 + 2] - pos[d * 3 + 2];
    const float r  = sqrtf(dx * dx + dy * dy + dz * dz + 1e-12f);

    float t = r * (1.0f / CUTOFF_R);
    float fc = 0.0f;
    if (t < 1.0f) {
        float den = fmaxf(1.0f - t * t, 1e-7f);
        fc = expf(1.0f - 1.0f / den);
    }
    float u = r / (r + 1.0f);
    u = fminf(fmaxf(u, 1e-7f), 1.0f - 1e-7f);
    const float lu = logf(u), l1mu = log1pf(-u);

    __syncthreads();   // logb + sW visible

    // each lane fills 16 of the 32 basis values for edge row ln
#pragma unroll
    for (int j = 0; j < 16; ++j) {
        int k = half * 16 + j;
        float kf = (float)k;
        sA[wave][ln * 32 + k] = expf(logb[k] + kf * lu + (31.0f - kf) * l1mu) * fc;
    }
    __syncthreads();   // rbf tile visible

    v8f acc0 = {}, acc1 = {};
    wmma_k32(sA[wave], sW, acc0, acc1);

    const int col = ln;
#pragma unroll
    for (int rr = 0; rr < 8; ++rr) {
        const int M = rr + half * 8;
        const int eM = wt * 16 + M;
        if (tvalid && eM < NEDGES) {
            const int sn = sSrc[wave][M], dn = sDst[wave][M];
            fatomic_add(&yacc[dn * 32 + col],      acc0[rr] * xf[sn * 32 + col]);
            fatomic_add(&yacc[dn * 32 + col + 16], acc1[rr] * xf[sn * 32 + col + 16]);
        }
    }
}

// ---------------------------------------------------------------------------
// Node update layer 0:  h = x0 + y;  h1 = silu(h@W1+b1);  h2 = h1@W2+b2;
// x1 = x0 + silu(c0)*h2.   One 16-node tile per wave.
// ---------------------------------------------------------------------------
__global__ void node_update0(const float* __restrict__ x0,
                             const float* __restrict__ yacc,
                             const float* __restrict__ W1, const float* __restrict__ b1,
                             const float* __restrict__ W2, const float* __restrict__ b2,
                             const float* __restrict__ c0p,
                             float* __restrict__ x1)
{
    __shared__ float sW1[1024], sW2[1024], sb1[32], sb2[32];
    __shared__ float sA[8][512], sB[8][512];

    const int tid = threadIdx.x;
    for (int i = tid; i < 1024; i += 256) { sW1[i] = W1[i]; sW2[i] = W2[i]; }
    if (tid < 32) { sb1[tid] = b1[tid]; sb2[tid] = b2[tid]; }

    const int wave = tid >> 5, lane = tid & 31;
    const int ln = lane & 15, half = lane >> 4, col = ln;

    const int numTiles = NNODES / 16;            // 3125 exactly
    const int tileId = blockIdx.x * 8 + wave;
    const bool valid = tileId < numTiles;
    const int tile = valid ? tileId : (numTiles - 1);
    const int n0 = tile * 16;

    const int nodeg = n0 + ln;
#pragma unroll
    for (int j = 0; j < 16; ++j) {
        int c = half * 16 + j;
        sA[wave][ln * 32 + c] = x0[nodeg * 32 + c] + yacc[nodeg * 32 + c];
    }
    __syncthreads();

    v8f a0 = {}, a1 = {};
    wmma_k32(sA[wave], sW1, a0, a1);
#pragma unroll
    for (int rr = 0; rr < 8; ++rr) {
        const int M = rr + half * 8;
        sB[wave][M * 32 + col]      = siluf(a0[rr] + sb1[col]);
        sB[wave][M * 32 + col + 16] = siluf(a1[rr] + sb1[col + 16]);
    }
    __syncthreads();

    v8f c0v = {}, c1v = {};
    wmma_k32(sB[wave], sW2, c0v, c1v);

    const float sc0 = siluf(c0p[0]);
    if (valid) {
#pragma unroll
        for (int rr = 0; rr < 8; ++rr) {
            const int ng = n0 + rr + half * 8;
            x1[ng * 32 + col]      = x0[ng * 32 + col]      + sc0 * (c0v[rr] + sb2[col]);
            x1[ng * 32 + col + 16] = x0[ng * 32 + col + 16] + sc0 * (c1v[rr] + sb2[col + 16]);
        }
    }
}

// ---------------------------------------------------------------------------
// Node update layer 1 + readout:
//   t = x1 + y0;  t1 = silu(t@W1+b1);  t2 = t1@W2+b2;  x2 = x1 + silu(c1)*t2;
//   h = silu(x2@Wro1+bro1);  e = h.Wro2 + bro2 + abias[z];  eg[seg[n]] += e
// ---------------------------------------------------------------------------
__global__ void node_update1(const float* __restrict__ x1,
                             const float* __restrict__ y0acc,
                             const float* __restrict__ W1, const float* __restrict__ b1,
                             const float* __restrict__ W2, const float* __restrict__ b2,
                             const float* __restrict__ c1p,
                             const float* __restrict__ Wro1, const float* __restrict__ bro1,
                             const float* __restrict__ Wro2, const float* __restrict__ bro2,
                             const float* __restrict__ abias,
                             const int*   __restrict__ z,
                             const int*   __restrict__ bseg,
                             float*       __restrict__ eg)
{
    __shared__ float sW1[1024], sW2[1024], sWro[1024];
    __shared__ float sb1[32], sb2[32], sbro[32], sWro2[32];
    __shared__ float sA[8][512], sB[8][512];

    const int tid = threadIdx.x;
    for (int i = tid; i < 1024; i += 256) { sW1[i] = W1[i]; sW2[i] = W2[i]; sWro[i] = Wro1[i]; }
    if (tid < 32) { sb1[tid] = b1[tid]; sb2[tid] = b2[tid]; sbro[tid] = bro1[tid]; sWro2[tid] = Wro2[tid]; }

    const int wave = tid >> 5, lane = tid & 31;
    const int ln = lane & 15, half = lane >> 4, col = ln;

    const int numTiles = NNODES / 16;
    const int tileId = blockIdx.x * 8 + wave;
    const bool valid = tileId < numTiles;
    const int tile = valid ? tileId : (numTiles - 1);
    const int n0 = tile * 16;

    const int nodeg = n0 + ln;
#pragma unroll
    for (int j = 0; j < 16; ++j) {
        int c = half * 16 + j;
        sA[wave][ln * 32 + c] = x1[nodeg * 32 + c] + y0acc[nodeg * 32 + c];
    }
    __syncthreads();

    v8f a0 = {}, a1 = {};
    wmma_k32(sA[wave], sW1, a0, a1);
#pragma unroll
    for (int rr = 0; rr < 8; ++rr) {
        const int M = rr + half * 8;
        sB[wave][M * 32 + col]      = siluf(a0[rr] + sb1[col]);
        sB[wave][M * 32 + col + 16] = siluf(a1[rr] + sb1[col + 16]);
    }
    __syncthreads();

    v8f d0 = {}, d1 = {};
    wmma_k32(sB[wave], sW2, d0, d1);
    const float sc1 = siluf(c1p[0]);
#pragma unroll
    for (int rr = 0; rr < 8; ++rr) {
        const int M = rr + half * 8;
        const int ng = n0 + M;
        sA[wave][M * 32 + col]      = x1[ng * 32 + col]      + sc1 * (d0[rr] + sb2[col]);
        sA[wave][M * 32 + col + 16] = x1[ng * 32 + col + 16] + sc1 * (d1[rr] + sb2[col + 16]);
    }
    __syncthreads();

    v8f h0 = {}, h1 = {};
    wmma_k32(sA[wave], sWro, h0, h1);
#pragma unroll
    for (int rr = 0; rr < 8; ++rr) {
        const int M = rr + half * 8;
        sB[wave][M * 32 + col]      = siluf(h0[rr] + sbro[col]);
        sB[wave][M * 32 + col + 16] = siluf(h1[rr] + sbro[col + 16]);
    }
    __syncthreads();

    if (valid && lane < 16) {
        const int n = n0 + lane;
        float e = bro2[0] + abias[z[n]];
#pragma unroll
        for (int f = 0; f < 32; ++f) e += sB[wave][lane * 32 + f] * sWro2[f];
        fatomic_add(&eg[bseg[n]], e);
    }
}

__global__ void finalize(const float* __restrict__ eg,
                         const unsigned char* __restrict__ mask,
                         float* __restrict__ out) {
    int g = blockIdx.x * 256 + threadIdx.x;
    if (g < NGRAPHS) out[g] = mask[g] ? eg[g] : 0.0f;
}

// ---------------------------------------------------------------------------
extern "C" void kernel_launch(void* const* d_in, const int* in_sizes, int n_in,
                              void* d_out, int out_size, void* d_ws, size_t ws_size,
                              hipStream_t stream) {
    const float* positions      = (const float*)d_in[0];
    const float* embed          = (const float*)d_in[1];
    const float* Wy0            = (const float*)d_in[2];
    const float* Wx0            = (const float*)d_in[3];
    const float* W1_0           = (const float*)d_in[4];
    const float* b1_0           = (const float*)d_in[5];
    const float* W2_0           = (const float*)d_in[6];
    const float* b2_0           = (const float*)d_in[7];
    const float* c0             = (const float*)d_in[8];
    const float* Wr_last        = (const float*)d_in[9];
    const float* W1_1           = (const float*)d_in[10];
    const float* b1_1           = (const float*)d_in[11];
    const float* W2_1           = (const float*)d_in[12];
    const float* b2_1           = (const float*)d_in[13];
    const float* c1             = (const float*)d_in[14];
    const float* Wro1           = (const float*)d_in[15];
    const float* bro1           = (const float*)d_in[16];
    const float* Wro2           = (const float*)d_in[17];
    const float* bro2           = (const float*)d_in[18];
    const float* atomic_bias    = (const float*)d_in[19];
    const int*   atomic_numbers = (const int*)d_in[20];
    const int*   dst_idx        = (const int*)d_in[21];
    const int*   src_idx        = (const int*)d_in[22];
    const int*   batch_segments = (const int*)d_in[23];
    const unsigned char* gmask  = (const unsigned char*)d_in[24];

    // Workspace layout (floats): x0 | x1 | wcomb | yacc | y0acc | eg
    float* ws    = (float*)d_ws;
    float* x0    = ws;
    float* x1    = x0 + NNODES * FDIM;
    float* wcomb = x1 + NNODES * FDIM;
    float* yacc  = wcomb + 1024;
    float* y0acc = yacc + NNODES * FDIM;
    float* eg    = y0acc + NNODES * FDIM;

    const int zspan = 2 * NNODES * FDIM + NGRAPHS;   // yacc..eg contiguous
    init_kernel<<<(zspan + 255) / 256, 256, 0, stream>>>(Wy0, Wx0, wcomb, yacc, zspan);
    gather_x0<<<(NNODES * FDIM + 255) / 256, 256, 0, stream>>>(embed, atomic_numbers, x0);

    const int edgeBlocks = (NEDGES + 127) / 128;   // 8 waves x 16 edges per block
    const int nodeBlocks = (NNODES + 127) / 128;   // 8 waves x 16 nodes per block

    edge_pass<<<edgeBlocks, 256, 0, stream>>>(positions, src_idx, dst_idx, x0, wcomb, yacc);
    node_update0<<<nodeBlocks, 256, 0, stream>>>(x0, yacc, W1_0, b1_0, W2_0, b2_0, c0, x1);
    edge_pass<<<edgeBlocks, 256, 0, stream>>>(positions, src_idx, dst_idx, x1, Wr_last, y0acc);
    node_update1<<<nodeBlocks, 256, 0, stream>>>(x1, y0acc, W1_1, b1_1, W2_1, b2_1, c1,
                                                 Wro1, bro1, Wro2, bro2, atomic_bias,
                                                 atomic_numbers, batch_segments, eg);
    finalize<<<(NGRAPHS + 255) / 256, 256, 0, stream>>>(eg, gmask, (float*)d_out);
}